// BRC_62715112457019
// MI455X (gfx1250) — compile-verified
//
#include <hip/hip_runtime.h>
#include <math.h>

typedef float v2f __attribute__((ext_vector_type(2)));
typedef float v8f __attribute__((ext_vector_type(8)));

#define N_  2
#define C_  64
#define H_  48
#define W_  48
#define HW_ (H_ * W_)     // 2304
#define NH_ 8             // heads
#define HD_ 8             // head dim
#define QT_ (HW_ / 16)    // 144 16-row tiles per (n, head)

__device__ __forceinline__ float sigmoidf_(float x) { return 1.0f / (1.0f + __expf(-x)); }

// ---------------------------------------------------------------------------
// Kernel 1: masks from sigmoid(P) + Sobel.  b_mask = edge>0, fg = Pm>0.5, bg = Pm<0.5
// ---------------------------------------------------------------------------
__global__ void masks_k(const float* __restrict__ P, float* __restrict__ biasfg,
                        float* __restrict__ bmask, float* __restrict__ fgm,
                        float* __restrict__ bbgm)
{
    int idx = blockIdx.x * blockDim.x + threadIdx.x;
    if (idx >= N_ * HW_) return;
    int n = idx / HW_, hw = idx % HW_;
    int y = hw / W_, x = hw % W_;
    const float* Pn = P + (size_t)n * HW_;
    float g[3][3];
    #pragma unroll
    for (int dy = -1; dy <= 1; ++dy)
        #pragma unroll
        for (int dx = -1; dx <= 1; ++dx) {
            int yy = y + dy, xx = x + dx;
            g[dy + 1][dx + 1] = (yy >= 0 && yy < H_ && xx >= 0 && xx < W_)
                                    ? sigmoidf_(Pn[yy * W_ + xx]) : 0.0f;
        }
    float gx = g[0][0] - g[0][2] + 2.f * (g[1][0] - g[1][2]) + g[2][0] - g[2][2];
    float gy = g[0][0] + 2.f * g[0][1] + g[0][2] - (g[2][0] + 2.f * g[2][1] + g[2][2]);
    float edge = sqrtf(gx * gx + gy * gy);
    float pm = g[1][1];
    float fg = (pm > 0.5f) ? 1.f : 0.f;
    float bg = (pm < 0.5f) ? 1.f : 0.f;
    float b  = (edge > 0.f) ? 1.f : 0.f;
    biasfg[idx] = (fg > 0.f) ? 0.f : -1e9f;
    bmask[idx]  = b;
    fgm[idx]    = fg;
    bbgm[idx]   = ((b + bg) > 0.f) ? 1.f : 0.f;
}

// ---------------------------------------------------------------------------
// Kernel 2: per-pixel LayerNorm over C, plus per-head l2-normalized q (=k)
// and raw v laid out row-major (hw, hd) for WMMA tile loads.
// ---------------------------------------------------------------------------
__global__ __launch_bounds__(128)
void ln_k(const float* __restrict__ F, const float* __restrict__ nw,
          const float* __restrict__ nb, float* __restrict__ Fn,
          float* __restrict__ qsp, float* __restrict__ vsp)
{
    int idx = blockIdx.x * blockDim.x + threadIdx.x;
    if (idx >= N_ * HW_) return;
    int n = idx / HW_, hw = idx % HW_;
    const float* Fb = F + (size_t)n * C_ * HW_ + hw;
    float v[C_];
    float mu = 0.f;
    #pragma unroll
    for (int c = 0; c < C_; ++c) { v[c] = Fb[(size_t)c * HW_]; mu += v[c]; }
    mu *= (1.0f / C_);
    float var = 0.f;
    #pragma unroll
    for (int c = 0; c < C_; ++c) { float d = v[c] - mu; var += d * d; }
    var *= (1.0f / C_);
    float inv = rsqrtf(var + 1e-5f);
    float* Fno = Fn + (size_t)n * C_ * HW_ + hw;
    #pragma unroll
    for (int c = 0; c < C_; ++c) {
        float f = (v[c] - mu) * inv * nw[c] + nb[c];
        v[c] = f;
        Fno[(size_t)c * HW_] = f;
    }
    #pragma unroll
    for (int h = 0; h < NH_; ++h) {
        float s2 = 0.f;
        #pragma unroll
        for (int d = 0; d < HD_; ++d) { float f = v[h * HD_ + d]; s2 += f * f; }
        float r = 1.0f / fmaxf(sqrtf(s2), 1e-12f);
        size_t base = ((size_t)(n * NH_ + h) * HW_ + hw) * HD_;
        #pragma unroll
        for (int d = 0; d < HD_; ++d) {
            qsp[base + d] = v[h * HD_ + d] * r;
            vsp[base + d] = v[h * HD_ + d];
        }
    }
}

// ---------------------------------------------------------------------------
// Kernel 3: spatial attention, one wave32 per (n, head, 16-query tile).
// Fixed-shift softmax: q,k are unit vectors -> logits = q.k + bias <= 1, so
// p = exp(s - 1) is overflow-safe with NO online max tracking and NO per-tile
// cross-lane reductions. Row sums accumulate per lane (column-partial) and are
// xor-reduced ONCE at the end. Inner loop: 2 WMMA (S=QK^T, hd=8=2*K4) +
// 8 exps + LDS-transpose of P + 4 WMMA (O+=P.V, 16 keys=4*K4).
// LDS ops are in-order within a wave -> transpose needs no barrier.
// ---------------------------------------------------------------------------
__global__ __launch_bounds__(256)
void spatial_k(const float* __restrict__ qsp, const float* __restrict__ vsp,
               const float* __restrict__ biasfg, float* __restrict__ spo)
{
    __shared__ float Pl[8][16][18];   // per-wave P tile; stride 18 => 8B-aligned v2f rows
    const int wave = threadIdx.x >> 5;
    const int lane = threadIdx.x & 31;
    const int tile = blockIdx.x * 8 + wave;      // grid sized exactly: no remainder
    const int qt = tile % QT_;
    const int nh = tile / QT_;                   // n*NH_ + h
    const float* qb = qsp + (size_t)nh * HW_ * HD_;
    const float* vb = vsp + (size_t)nh * HW_ * HD_;
    const float* bs = biasfg + (size_t)(nh / NH_) * HW_;
    const int half = lane >> 4;                  // 0/1: K pair group (f32 A/B layout)
    const int l16  = lane & 15;
    const int q0 = qt * 16;

    // A tiles (Q rows): lane l holds row q0+l16, K elems {2*half, 2*half+1}
    v2f a0 = *(const v2f*)(qb + (size_t)(q0 + l16) * HD_ + 2 * half);       // dims 0..3
    v2f a1 = *(const v2f*)(qb + (size_t)(q0 + l16) * HD_ + 4 + 2 * half);   // dims 4..7

    v8f O = {0.f, 0.f, 0.f, 0.f, 0.f, 0.f, 0.f, 0.f};
    float prow[8];                               // per-lane partial row sums
    #pragma unroll
    for (int r = 0; r < 8; ++r) prow[r] = 0.f;

    for (int kt = 0; kt < QT_; ++kt) {
        const int k0 = kt * 16;
        // B tiles (K^T): B(k,n) = K[k0+n][k]; K == Q (normalized feats)
        v2f b0 = *(const v2f*)(qb + (size_t)(k0 + l16) * HD_ + 2 * half);
        v2f b1 = *(const v2f*)(qb + (size_t)(k0 + l16) * HD_ + 4 + 2 * half);
        if (kt + 1 < QT_) {                      // warm L0/GL2 for the next key tile
            __builtin_prefetch(qb + (size_t)(k0 + 16 + l16) * HD_, 0, 3);
            __builtin_prefetch(vb + (size_t)(k0 + 16 + l16) * HD_, 0, 3);
        }
        v8f S = {0.f, 0.f, 0.f, 0.f, 0.f, 0.f, 0.f, 0.f};
        S = __builtin_amdgcn_wmma_f32_16x16x4_f32(false, a0, false, b0, (short)0, S, false, false);
        S = __builtin_amdgcn_wmma_f32_16x16x4_f32(false, a1, false, b1, (short)0, S, false, false);

        const float kb = bs[k0 + l16] - 1.0f;    // key bias + fixed softmax shift
        #pragma unroll
        for (int r = 0; r < 8; ++r) {            // lane holds (m = r+8*half, n = l16)
            float pe = __expf(S[r] + kb);        // <= 1 always; masked keys -> 0
            prow[r] += pe;
            Pl[wave][r + 8 * half][l16] = pe;    // stage P (D layout) -> LDS
        }
        // re-read P as A layout and accumulate O += P.V (4 chained K4 WMMAs)
        #pragma unroll
        for (int kc = 0; kc < 4; ++kc) {
            v2f pa = *(const v2f*)&Pl[wave][l16][kc * 4 + 2 * half];
            const int vr = k0 + kc * 4 + 2 * half;
            const int d  = l16 & 7;              // cols 8..15 duplicate dims (unused)
            v2f vbt;
            vbt.x = vb[(size_t)(vr + 0) * HD_ + d];
            vbt.y = vb[(size_t)(vr + 1) * HD_ + d];
            O = __builtin_amdgcn_wmma_f32_16x16x4_f32(false, pa, false, vbt, (short)0, O, false, false);
        }
    }

    // single end-of-loop row-sum reduction across the 16-lane group
    #pragma unroll
    for (int r = 0; r < 8; ++r) {
        float rs = prow[r];
        rs += __shfl_xor(rs, 1);
        rs += __shfl_xor(rs, 2);
        rs += __shfl_xor(rs, 4);
        rs += __shfl_xor(rs, 8);
        prow[r] = fmaxf(rs, 1e-30f);
    }

    if (l16 < HD_) {
        #pragma unroll
        for (int r = 0; r < 8; ++r) {
            const int m = r + 8 * half;
            const float q = qb[(size_t)(q0 + m) * HD_ + l16];
            spo[((size_t)nh * HW_ + (q0 + m)) * HD_ + l16] = O[r] / prow[r] + q;
        }
    }
}

// ---------------------------------------------------------------------------
// Kernel 4: channel attention (8x8 Gram over HW=2304), one block per (n, head)
// ---------------------------------------------------------------------------
__global__ __launch_bounds__(256)
void chan_k(const float* __restrict__ Fn, const float* __restrict__ fgm,
            const float* __restrict__ bbgm, float* __restrict__ cho)
{
    const int nh = blockIdx.x;
    const int n = nh / NH_, h = nh % NH_;
    const float* rows = Fn + ((size_t)n * C_ + h * HD_) * HW_;
    const float* fg = fgm + (size_t)n * HW_;
    const float* bb = bbgm + (size_t)n * HW_;
    float Macc[8][8], nq2[8], nk2[8];
    #pragma unroll
    for (int q = 0; q < 8; ++q) {
        nq2[q] = 0.f; nk2[q] = 0.f;
        #pragma unroll
        for (int k = 0; k < 8; ++k) Macc[q][k] = 0.f;
    }
    for (int s = threadIdx.x; s < HW_; s += 256) {
        const float fgv = fg[s], bbv = bb[s];
        float a[8], kv[8];
        #pragma unroll
        for (int i = 0; i < 8; ++i) {
            const float f = rows[(size_t)i * HW_ + s];
            a[i] = f * bbv; kv[i] = f * fgv;
        }
        #pragma unroll
        for (int q = 0; q < 8; ++q) {
            nq2[q] += a[q] * a[q];
            nk2[q] += kv[q] * kv[q];
            #pragma unroll
            for (int k = 0; k < 8; ++k) Macc[q][k] += a[q] * kv[k];
        }
    }
    __shared__ float sacc[80];
    if (threadIdx.x < 80) sacc[threadIdx.x] = 0.f;
    __syncthreads();
    #pragma unroll
    for (int q = 0; q < 8; ++q) {
        #pragma unroll
        for (int k = 0; k < 8; ++k) atomicAdd(&sacc[q * 8 + k], Macc[q][k]);
        atomicAdd(&sacc[64 + q], nq2[q]);
        atomicAdd(&sacc[72 + q], nk2[q]);
    }
    __syncthreads();
    float rnq[8], rnk[8];
    #pragma unroll
    for (int i = 0; i < 8; ++i) {
        rnq[i] = 1.0f / fmaxf(sqrtf(sacc[64 + i]), 1e-12f);
        rnk[i] = 1.0f / fmaxf(sqrtf(sacc[72 + i]), 1e-12f);
    }
    float attn[8][8];
    #pragma unroll
    for (int q = 0; q < 8; ++q) {
        float lg[8]; float mx = -1e30f;
        #pragma unroll
        for (int k = 0; k < 8; ++k) { lg[k] = sacc[q * 8 + k] * rnq[q] * rnk[k]; mx = fmaxf(mx, lg[k]); }
        float sm = 0.f;
        #pragma unroll
        for (int k = 0; k < 8; ++k) { lg[k] = __expf(lg[k] - mx); sm += lg[k]; }
        const float inv = 1.0f / sm;
        #pragma unroll
        for (int k = 0; k < 8; ++k) attn[q][k] = lg[k] * inv;
    }
    float* co = cho + ((size_t)n * C_ + h * HD_) * HW_;
    for (int s = threadIdx.x; s < HW_; s += 256) {
        const float fgv = fg[s], bbv = bb[s];
        float kv[8];
        #pragma unroll
        for (int k = 0; k < 8; ++k) kv[k] = rows[(size_t)k * HW_ + s] * fgv;
        #pragma unroll
        for (int q = 0; q < 8; ++q) {
            float o = rows[(size_t)q * HW_ + s] * bbv * rnq[q];
            #pragma unroll
            for (int k = 0; k < 8; ++k) o += attn[q][k] * kv[k];
            co[(size_t)q * HW_ + s] = o;
        }
    }
}

// ---------------------------------------------------------------------------
// Kernel 5: out = Fn + (b_mask ? spatial_out : Fn) + channel_out
// ---------------------------------------------------------------------------
__global__ void combine_k(const float* __restrict__ Fn, const float* __restrict__ spo,
                          const float* __restrict__ cho, const float* __restrict__ bm,
                          float* __restrict__ out)
{
    const int idx = blockIdx.x * blockDim.x + threadIdx.x;
    if (idx >= N_ * C_ * HW_) return;
    const int n  = idx / (C_ * HW_);
    const int c  = (idx / HW_) % C_;
    const int hw = idx % HW_;
    const float fn = Fn[idx];
    const float ch = cho[idx];
    const float b  = bm[n * HW_ + hw];
    float sp = fn;
    if (b > 0.f)
        sp = spo[((size_t)(n * NH_ + c / HD_) * HW_ + hw) * HD_ + (c % HD_)];
    out[idx] = fn + sp + ch;
}

// ---------------------------------------------------------------------------
extern "C" void kernel_launch(void* const* d_in, const int* in_sizes, int n_in,
                              void* d_out, int out_size, void* d_ws, size_t ws_size,
                              hipStream_t stream)
{
    (void)in_sizes; (void)n_in; (void)out_size; (void)ws_size;
    const float* F  = (const float*)d_in[0];
    const float* P  = (const float*)d_in[1];
    const float* nw = (const float*)d_in[2];
    const float* nb = (const float*)d_in[3];
    float* out = (float*)d_out;
    float* ws  = (float*)d_ws;

    const size_t SZ = (size_t)N_ * C_ * HW_;   // 294912
    float* Fn   = ws;
    float* qsp  = ws + SZ;
    float* vsp  = ws + 2 * SZ;
    float* spo  = ws + 3 * SZ;
    float* cho  = ws + 4 * SZ;
    float* bias = ws + 5 * SZ;
    float* bm   = bias + (size_t)N_ * HW_;
    float* fgm  = bm   + (size_t)N_ * HW_;
    float* bbgm = fgm  + (size_t)N_ * HW_;

    masks_k<<<(N_ * HW_ + 255) / 256, 256, 0, stream>>>(P, bias, bm, fgm, bbgm);
    ln_k<<<(N_ * HW_ + 127) / 128, 128, 0, stream>>>(F, nw, nb, Fn, qsp, vsp);
    // 2 * 8 * 144 = 2304 wave-tiles, 8 waves (256 thr) per block -> exactly 288 blocks
    spatial_k<<<(N_ * NH_ * QT_) / 8, 256, 0, stream>>>(qsp, vsp, bias, spo);
    chan_k<<<N_ * NH_, 256, 0, stream>>>(Fn, fgm, bbgm, cho);
    combine_k<<<(int)((SZ + 255) / 256), 256, 0, stream>>>(Fn, spo, cho, bm, out);
}